// StyleTransferVectorQuantizer_26654567039065
// MI455X (gfx1250) — compile-verified
//
#include <hip/hip_runtime.h>
#include <hip/hip_bf16.h>

typedef float v2f __attribute__((ext_vector_type(2)));
typedef float v8f __attribute__((ext_vector_type(8)));

#define N_E     1024
#define E_DIM   256
#define T_DIM   4096
#define N_TOK   65536
#define ZQ_ELEMS 16777216  // 16*256*4096

// output layout (floats), concatenated in reference return order
constexpr size_t OFF_ZQ     = 0;
constexpr size_t OFF_LOSS   = 16777216;
constexpr size_t OFF_PERP   = 16777217;
constexpr size_t OFF_ONEHOT = 16777218;
constexpr size_t OFF_IDX    = 16777218ull + (size_t)N_TOK * N_E;  // 83886082

// LDS swizzle: XOR channel index by (row&15)<<2 — keeps 8B pairs intact,
// spreads the 16 A-fragment lanes across all 64 banks.
__device__ __forceinline__ int zswz(int row, int k) {
    return row * E_DIM + (k ^ ((row & 15) << 2));
}

// ---------------------------------------------------------------- prep ----
__global__ __launch_bounds__(256) void vq_prep_kernel(
    const float* __restrict__ emb, const float* __restrict__ pos,
    const float* __restrict__ neg, const float* __restrict__ style_tok,
    float* __restrict__ new_w, float* __restrict__ wnorm,
    float* __restrict__ counts, double* __restrict__ loss_acc,
    unsigned long long* __restrict__ winner) {
    __shared__ float red[256];
    const int j = blockIdx.x;          // codebook row
    const int k = threadIdx.x;         // channel
    const float s = style_tok[0];
    const size_t e = (size_t)j * E_DIM + k;
    const float w = emb[e] * (s * pos[e] + (1.0f - s) * neg[e]);
    new_w[e] = w;
    red[k] = w * w;
    __syncthreads();
    for (int off = 128; off > 0; off >>= 1) {
        if (k < off) red[k] += red[k + off];
        __syncthreads();
    }
    if (k == 0) { wnorm[j] = red[0]; counts[j] = 0.0f; }
    if (k < 64) winner[(size_t)blockIdx.x * 64 + k] = 0xFFFFFFFFFFFFFFFFull;
    if (blockIdx.x == 0 && k == 0) *loss_acc = 0.0;
}

// ---------------------------------------------------------------- zero ----
__global__ void vq_zero_kernel(float4* __restrict__ p, long n4) {
    long i = (long)blockIdx.x * blockDim.x + threadIdx.x;
    const long stride = (long)gridDim.x * blockDim.x;
    const float4 z = make_float4(0.f, 0.f, 0.f, 0.f);
    for (; i < n4; i += stride) p[i] = z;
}

// ------------------------------------------------- fused GEMM + argmin ----
// block = 64 tokens x full codebook; wave w owns codebook cols [w*128, w*128+128)
__global__ __launch_bounds__(256) void vq_gemm_argmin_kernel(
    const float* __restrict__ z, const float* __restrict__ new_w,
    const float* __restrict__ wnorm, unsigned long long* __restrict__ winner) {
    __shared__ float zt[64 * E_DIM];   // 64 KB, swizzled [row][k]

    const int m0  = blockIdx.x * 64;
    const int b   = m0 / T_DIM;
    const int t0  = m0 % T_DIM;        // tiles of 64 never straddle a batch
    const int tid = threadIdx.x;

    // stage z tile: thread tid = channel, 64 consecutive t (coalesced float4)
    {
        const float* src = z + ((size_t)(b * E_DIM + tid)) * T_DIM + t0;
        for (int r = 0; r < 64; r += 4) {
            float4 v = *(const float4*)(src + r);
            zt[zswz(r + 0, tid)] = v.x;
            zt[zswz(r + 1, tid)] = v.y;
            zt[zswz(r + 2, tid)] = v.z;
            zt[zswz(r + 3, tid)] = v.w;
        }
    }
    __syncthreads();

    const int wave  = tid >> 5;
    const int lane  = tid & 31;
    const int mres  = lane & 15;        // A: M = lane&15 ; B/C: N = lane&15
    const int nhalf = lane >> 4;        // C: M += 8 on upper half
    const int kh    = nhalf << 1;       // A/B: upper lanes hold K+2,K+3
    const int ncol0 = wave * 128;

    float runmin[4][8];
    int   runidx[4][8];
#pragma unroll
    for (int mt = 0; mt < 4; ++mt)
#pragma unroll
        for (int v = 0; v < 8; ++v) { runmin[mt][v] = __builtin_inff(); runidx[mt][v] = 0; }

    for (int nt = 0; nt < 8; ++nt) {
        const int ncol = ncol0 + nt * 16 + mres;               // this lane's column
        const float* wptr = new_w + (size_t)ncol * E_DIM + kh; // B fragment base
        // warm L0/L2 for the next N-tile's codebook rows (global_prefetch_b8)
        if (nt + 1 < 8) __builtin_prefetch(wptr + (size_t)16 * E_DIM, 0, 3);
        const float wn = wnorm[ncol];                          // hoisted: overlaps GEMM

        v8f accs[4];
#pragma unroll
        for (int mt = 0; mt < 4; ++mt) accs[mt] = (v8f){0.f,0.f,0.f,0.f,0.f,0.f,0.f,0.f};

        // software pipeline: B prefetched 2 k-steps ahead (L2 latency),
        // A prefetched 1 k-step ahead (LDS latency)
        v2f b0 = *(const v2f*)(wptr);
        v2f b1 = *(const v2f*)(wptr + 4);
        v2f anxt[4];
#pragma unroll
        for (int mt = 0; mt < 4; ++mt)
            anxt[mt] = *(const v2f*)(&zt[zswz(mt * 16 + mres, kh)]);

        for (int k0 = 0; k0 < E_DIM; k0 += 4) {
            const v2f bcur = b0;
            b0 = b1;
            v2f acur[4];
#pragma unroll
            for (int mt = 0; mt < 4; ++mt) acur[mt] = anxt[mt];

            const int kn  = (k0 + 4) & (E_DIM - 1);  // wrap: harmless reload
            const int kn2 = (k0 + 8) & (E_DIM - 1);
            b1 = *(const v2f*)(wptr + kn2);
#pragma unroll
            for (int mt = 0; mt < 4; ++mt)
                anxt[mt] = *(const v2f*)(&zt[zswz(mt * 16 + mres, kn + kh)]);

#pragma unroll
            for (int mt = 0; mt < 4; ++mt)
                accs[mt] = __builtin_amdgcn_wmma_f32_16x16x4_f32(
                    false, acur[mt], false, bcur, (short)0, accs[mt], false, false);
        }

#pragma unroll
        for (int mt = 0; mt < 4; ++mt)
#pragma unroll
            for (int v = 0; v < 8; ++v) {
                const float score = wn - 2.0f * accs[mt][v];   // ||w||^2 - 2 z.w
                if (score < runmin[mt][v]) { runmin[mt][v] = score; runidx[mt][v] = ncol; }
            }
    }

    // reduce across the 16 lanes of each half (they hold the same M rows)
#pragma unroll
    for (int mt = 0; mt < 4; ++mt)
#pragma unroll
        for (int v = 0; v < 8; ++v) {
            float val = runmin[mt][v];
            int   idx = runidx[mt][v];
#pragma unroll
            for (int m = 8; m >= 1; m >>= 1) {
                const float ov = __shfl_xor(val, m, 32);
                const int   oi = __shfl_xor(idx, m, 32);
                if (ov < val || (ov == val && oi < idx)) { val = ov; idx = oi; }
            }
            if (mres == 0) {  // lanes 0 and 16 hold winners for rows v / v+8
                const int row = m0 + mt * 16 + v + (nhalf << 3);
                const unsigned int bu = __float_as_uint(val);
                const unsigned int u  = (bu & 0x80000000u) ? ~bu : (bu | 0x80000000u);
                const unsigned long long packed =
                    ((unsigned long long)u << 32) | (unsigned int)idx;
                atomicMin(&winner[row], packed);
            }
        }
}

// -------------------------------------------------------------- gather ----
__global__ __launch_bounds__(256) void vq_gather_kernel(
    const unsigned long long* __restrict__ winner,
    const float* __restrict__ new_w, float* __restrict__ out,
    float* __restrict__ counts) {
    __shared__ int sidx[64];
    const int m0  = blockIdx.x * 64;
    const int tid = threadIdx.x;
    if (tid < 64) {
        const int n = m0 + tid;
        const int idx = (int)(winner[n] & 0xFFFFFFFFull);
        sidx[tid] = idx;
        out[OFF_IDX + n] = (float)idx;                       // indices (as f32)
        out[OFF_ONEHOT + (size_t)n * N_E + idx] = 1.0f;      // one-hot scatter
        atomicAdd(&counts[idx], 1.0f);
    }
    __syncthreads();
    const int b  = m0 / T_DIM, t0 = m0 % T_DIM;
    const int c  = tid;
    float* dst = out + OFF_ZQ + ((size_t)(b * E_DIM + c)) * T_DIM + t0;
    for (int r = 0; r < 64; r += 4) {
        float4 v;
        v.x = new_w[(size_t)sidx[r + 0] * E_DIM + c];
        v.y = new_w[(size_t)sidx[r + 1] * E_DIM + c];
        v.z = new_w[(size_t)sidx[r + 2] * E_DIM + c];
        v.w = new_w[(size_t)sidx[r + 3] * E_DIM + c];
        *(float4*)(dst + r) = v;   // 16B-aligned: t0 % 64 == 0
    }
}

// ---------------------------------------------------------------- loss ----
__global__ __launch_bounds__(256) void vq_loss_kernel(
    const float* __restrict__ z, const float* __restrict__ zq,
    double* __restrict__ loss_acc) {
    __shared__ double red[256];
    size_t i = (size_t)blockIdx.x * blockDim.x + threadIdx.x;
    const size_t stride = (size_t)gridDim.x * blockDim.x;
    double s = 0.0;
    for (; i < (size_t)ZQ_ELEMS; i += stride) {
        const float d = zq[i] - z[i];
        s += (double)(d * d);
    }
    red[threadIdx.x] = s;
    __syncthreads();
    for (int off = 128; off > 0; off >>= 1) {
        if (threadIdx.x < off) red[threadIdx.x] += red[threadIdx.x + off];
        __syncthreads();
    }
    if (threadIdx.x == 0) atomicAdd(loss_acc, red[0]);
}

// ------------------------------------------------------------ finalize ----
__global__ __launch_bounds__(1024) void vq_finalize_kernel(
    const float* __restrict__ counts, const double* __restrict__ loss_acc,
    float* __restrict__ out) {
    __shared__ float red[1024];
    const int t = threadIdx.x;
    const float e = counts[t] / (float)N_TOK;
    red[t] = e * logf(e + 1e-10f);
    __syncthreads();
    for (int off = 512; off > 0; off >>= 1) {
        if (t < off) red[t] += red[t + off];
        __syncthreads();
    }
    if (t == 0) {
        out[OFF_PERP] = expf(-red[0]);
        out[OFF_LOSS] = (float)(1.25 * (*loss_acc) / (double)ZQ_ELEMS);
    }
}

// ---------------------------------------------------------------- host ----
extern "C" void kernel_launch(void* const* d_in, const int* in_sizes, int n_in,
                              void* d_out, int out_size, void* d_ws, size_t ws_size,
                              hipStream_t stream) {
    const float* z     = (const float*)d_in[0];  // (16,256,4096)
    const float* style = (const float*)d_in[1];  // (1,1)
    const float* emb   = (const float*)d_in[2];  // (1024,256)
    const float* pos   = (const float*)d_in[3];  // (1024,256)
    const float* neg   = (const float*)d_in[4];  // (1024,256)
    float* out = (float*)d_out;

    char* ws = (char*)d_ws;
    float* new_w  = (float*)ws;                                   // 1 MB
    float* wnorm  = (float*)(ws + (size_t)N_E * E_DIM * 4);       // 4 KB
    float* counts = wnorm + N_E;                                  // 4 KB
    double* loss_acc = (double*)(counts + N_E);                   // 8 B (8-aligned)
    unsigned long long* winner = (unsigned long long*)(loss_acc + 1); // 512 KB

    vq_prep_kernel<<<N_E, 256, 0, stream>>>(emb, pos, neg, style,
                                            new_w, wnorm, counts, loss_acc, winner);
    // zero [OFF_LOSS, OFF_LOSS + 4*16777217) floats: covers loss/perp + one-hot
    vq_zero_kernel<<<2048, 256, 0, stream>>>((float4*)(out + OFF_LOSS), 16777217L);
    vq_gemm_argmin_kernel<<<N_TOK / 64, 256, 0, stream>>>(z, new_w, wnorm, winner);
    vq_gather_kernel<<<N_TOK / 64, 256, 0, stream>>>(winner, new_w, out, counts);
    vq_loss_kernel<<<1024, 256, 0, stream>>>(z, out + OFF_ZQ, loss_acc);
    vq_finalize_kernel<<<1, 1024, 0, stream>>>(counts, loss_acc, out);
    (void)in_sizes; (void)n_in; (void)out_size; (void)ws_size;
}